// BasisPooling_82849919140401
// MI455X (gfx1250) — compile-verified
//
#include <hip/hip_runtime.h>

// BasisPooling on gfx1250:
//   out[b, 2c+d, oy, ox] = sum_k W[d,k] * patch[b,c,k, oy,ox]
// W[d,k] = softmax_k(pos[k]*off[d]/T) is data-independent (softmax shift-
// invariance cancels the variance term). Applied via V_WMMA_F32_16X16X4_F32,
// 16 patches per wave-WMMA:
//   B (4x16): one global_load_b64 per wave lands patch data directly in the
//             native WMMA B layout (lanes 0-15 = top-row pairs, lanes 16-31 =
//             bottom-row pairs).
//   A (16x4): row 0 = W0, row 8 = W1, rest 0  ==>  D VGPR0 holds channel 2c
//             in lanes 0-15 and channel 2c+1 in lanes 16-31: one full-wave
//             NT b32 store writes both output channels.
// Software-pipelined 4 deep so each wave keeps 4 x 256B loads in flight
// (needed to reach the 23.3 TB/s roofline, ~13 us for 308 MB of traffic).

typedef float v2f __attribute__((ext_vector_type(2)));
typedef float v8f __attribute__((ext_vector_type(8)));

__global__ __launch_bounds__(256) void BasisPooling_82849919140401_kernel(
    const float* __restrict__ x,            // [B, C, 112, 112]
    const float* __restrict__ temperature,  // [1]
    float* __restrict__ out)                // [B, 2C, 56, 56]
{
    constexpr int W  = 112;
    constexpr int OW = 56;
    constexpr int L  = 56 * 56;    // 3136
    constexpr int NSTRIP = L / 16; // 196 strips of 16 patches (exact)

    const int lane  = threadIdx.x & 31;
    const int wave  = threadIdx.x >> 5;   // 0..7
    const int m     = lane & 15;          // patch index within strip / A row
    const int khalf = lane >> 4;          // 0: k={0,1}, 1: k={2,3}

    // ---- Build A = [16x4] with row0 = W0 (off=-0.5), row8 = W1 (off=+0.5) ----
    const float T   = temperature[0];
    const float off = (m == 8) ? 0.5f : -0.5f;
    const float s   = off / (3.0f * T);          // score step: pos_k = k/3
    const float mx  = fmaxf(0.0f, 3.0f * s);     // max of {0,s,2s,3s}
    const float e0  = __expf(0.0f   - mx);
    const float e1  = __expf(s      - mx);
    const float e2  = __expf(2.0f*s - mx);
    const float e3  = __expf(3.0f*s - mx);
    const float inv = 1.0f / (e0 + e1 + e2 + e3);
    const bool  act = (m == 0) || (m == 8);
    v2f a;
    a.x = act ? (khalf ? e2 : e0) * inv : 0.0f;  // A[m, 2*khalf]
    a.y = act ? (khalf ? e3 : e1) * inv : 0.0f;  // A[m, 2*khalf+1]

    // One (b,c) input plane per block; two output channel planes.
    const float* __restrict__ xin = x + (size_t)blockIdx.x * (size_t)(112 * 112);
    float* __restrict__ obase     = out + (size_t)blockIdx.x * (size_t)(2 * L);

    // Strips owned by this wave: wave, wave+8, ... (< 196). Wave-uniform count.
    const int nmine = (NSTRIP - 1 - wave) / 8 + 1;   // 25 (waves 0-3) or 24

    int i = 0;
    // ---- Main pipelined body: 4 loads in flight, then 4 WMMA + 4 stores ----
    for (; i + 4 <= nmine; i += 4) {
        v2f bm[4];
        int lidx[4];
        #pragma unroll
        for (int j = 0; j < 4; ++j) {
            const int l  = (wave + 8 * (i + j)) * 16 + m;
            lidx[j] = l;
            const int oy = (int)((unsigned)l / OW);
            const int ox = l - oy * OW;
            const int iy = 2 * oy + khalf;
            bm[j] = *(const v2f*)(xin + iy * W + 2 * ox);  // global_load_b64
        }
        #pragma unroll
        for (int j = 0; j < 4; ++j) {
            v8f acc = {};
            acc = __builtin_amdgcn_wmma_f32_16x16x4_f32(
                false, a, false, bm[j], (short)0, acc, false, false);
            __builtin_nontemporal_store(acc[0], obase + khalf * L + lidx[j]);
        }
    }
    // ---- Tail (at most 3 strips; wave-uniform, EXEC stays all-ones) ----
    for (; i < nmine; ++i) {
        const int l  = (wave + 8 * i) * 16 + m;
        const int oy = (int)((unsigned)l / OW);
        const int ox = l - oy * OW;
        const int iy = 2 * oy + khalf;
        v2f bm = *(const v2f*)(xin + iy * W + 2 * ox);
        v8f acc = {};
        acc = __builtin_amdgcn_wmma_f32_16x16x4_f32(
            false, a, false, bm, (short)0, acc, false, false);
        __builtin_nontemporal_store(acc[0], obase + khalf * L + l);
    }
}

extern "C" void kernel_launch(void* const* d_in, const int* in_sizes, int n_in,
                              void* d_out, int out_size, void* d_ws, size_t ws_size,
                              hipStream_t stream) {
    const float* x    = (const float*)d_in[0];
    const float* temp = (const float*)d_in[1];
    float* out        = (float*)d_out;

    const int planes = in_sizes[0] / (112 * 112);  // B*C = 4096
    BasisPooling_82849919140401_kernel<<<planes, 256, 0, stream>>>(x, temp, out);
}